// TopoLoss_77189152244470
// MI455X (gfx1250) — compile-verified
//
#include <hip/hip_runtime.h>
#include <math.h>

typedef float v2f __attribute__((ext_vector_type(2)));
typedef float v8f __attribute__((ext_vector_type(8)));

#define N_PTS   1024
#define NT      256
#define N_BATCH 16
#define INF_F   1e30f
#define EPS_F   1e-12f

#if defined(__has_builtin)
#if __has_builtin(__builtin_amdgcn_wmma_f32_16x16x4_f32)
#define HAVE_WMMA_F32 1
#endif
#if __has_builtin(__builtin_amdgcn_mov_dpp8)
#define HAVE_DPP8 1
#endif
#endif

// DPP8 selector: lane l reads lane sel[l] within its 8-lane group.
#define DPP8_SEL(a,b,c,d,e,f,g,h) \
  ((a) | ((b) << 3) | ((c) << 6) | ((d) << 9) | ((e) << 12) | ((f) << 15) | \
   ((g) << 18) | ((h) << 21))
#define DPP8_XOR1 DPP8_SEL(1, 0, 3, 2, 5, 4, 7, 6)
#define DPP8_XOR2 DPP8_SEL(2, 3, 0, 1, 6, 7, 4, 5)
#define DPP8_XOR4 DPP8_SEL(4, 5, 6, 7, 0, 1, 2, 3)

struct PrimLds {
  v2f a0[N_PTS];               // per point: (-2x, -2y)
  v2f a1[N_PTS];               // per point: (-2z, |p|^2)
  unsigned long long slot[4];  // rotating argmin slots (u64 keys)
  float pval[8];               // block_sum scratch
};

// Prim's MST on squared Euclidean distances of one 1024x3 cloud; leaves the
// 1023 MST-edge d^2 values ascending-sorted in deaths[0..1022] (deaths[1023]
// = INF pad). Block = 256 threads = 8 waves (wave32).
__device__ __forceinline__ void prim_sort_d2(const float* __restrict__ pts,
                                             PrimLds& L,
                                             float* deaths /* 1024 floats in LDS */) {
  const int tid = threadIdx.x;
  __syncthreads();
  for (int i = tid; i < N_PTS; i += NT) {
    float px = pts[3 * i + 0];
    float py = pts[3 * i + 1];
    float pz = pts[3 * i + 2];
    v2f p0, p1;
    p0[0] = -2.0f * px;
    p0[1] = -2.0f * py;
    p1[0] = -2.0f * pz;
    p1[1] = px * px + py * py + pz * pz;
    L.a0[i] = p0;
    L.a1[i] = p1;
  }
  if (tid < 4) L.slot[tid] = ~0ull;
  __syncthreads();

  const int  wave = tid >> 5;
  const int  lane = tid & 31;
  const int  hi   = lane >> 4;   // lane-half selects K group {0,1} vs {2,3}
  const bool hiB  = (hi != 0);
  const int  col  = lane & 15;   // N (column) within a 16-wide tile

  // Per step each lane needs exactly one A-pair: K={0,1} -> a0, K={2,3} -> a1.
  const v2f* __restrict__ apA = hiB ? L.a1 : L.a0;

  // B operand tiles: wave owns columns [wave*128, wave*128+128) as 8 tiles.
  // B[K][N] with K = vgpr + 2*hi, N = col.  b = (x, y, z, 1).
  v2f Bt[8];
#pragma unroll
  for (int t = 0; t < 8; ++t) {
    int q = wave * 128 + t * 16 + col;
    v2f p0 = L.a0[q];
    v2f p1 = L.a1[q];
    float qx = -0.5f * p0[0];   // exact
    float qy = -0.5f * p0[1];
    float qz = -0.5f * p1[0];
    Bt[t][0] = hiB ? qz   : qx;  // K = 2*hi
    Bt[t][1] = hiB ? 1.0f : qy;  // K = 2*hi + 1
  }

  // Each lane owns 4 nodes (output tile t = 2*s + hi, column = col).
  int   node_[4];
  float nq_[4];
  float mind_[4];
#if !HAVE_WMMA_F32
  float qx_[4], qy_[4], qz_[4];
#endif
  int vis = 0;
#pragma unroll
  for (int s = 0; s < 4; ++s) {
    int n = wave * 128 + (2 * s + hi) * 16 + col;
    node_[s] = n;
    nq_[s] = L.a1[n][1];
    mind_[s] = INF_F;
    vis |= (int)(n == 0) << s;  // node 0 starts the tree
#if !HAVE_WMMA_F32
    qx_[s] = -0.5f * L.a0[n][0];
    qy_[s] = -0.5f * L.a0[n][1];
    qz_[s] = -0.5f * L.a1[n][0];
#endif
  }

  v2f Aj = apA[0];  // A operand of current tree node j (j = 0 initially)
#if !HAVE_WMMA_F32
  v2f j0 = L.a0[0], j1 = L.a1[0];
#endif

  for (int step = 0; step < N_PTS - 1; ++step) {
    float d2o[4];
#if HAVE_WMMA_F32
    // A matrix: all 16 M-rows identical = [-2jx, -2jy, -2jz, |pj|^2]
    v8f cz = {0.0f, 0.0f, 0.0f, 0.0f, 0.0f, 0.0f, 0.0f, 0.0f};
#define WMMA_D2(BV) __builtin_amdgcn_wmma_f32_16x16x4_f32( \
    false, Aj, false, (BV), (short)0, cz, false, false)
    v8f c0 = WMMA_D2(Bt[0]);
    v8f c1 = WMMA_D2(Bt[1]);
    v8f c2 = WMMA_D2(Bt[2]);
    v8f c3 = WMMA_D2(Bt[3]);
    v8f c4 = WMMA_D2(Bt[4]);
    v8f c5 = WMMA_D2(Bt[5]);
    v8f c6 = WMMA_D2(Bt[6]);
    v8f c7 = WMMA_D2(Bt[7]);
#undef WMMA_D2
    // All rows identical -> element 0 is fine; tile = 2*s + hi.
    d2o[0] = (hiB ? c1[0] : c0[0]) + nq_[0];
    d2o[1] = (hiB ? c3[0] : c2[0]) + nq_[1];
    d2o[2] = (hiB ? c5[0] : c4[0]) + nq_[2];
    d2o[3] = (hiB ? c7[0] : c6[0]) + nq_[3];
#else
#pragma unroll
    for (int s = 0; s < 4; ++s)
      d2o[s] = j0[0] * qx_[s] + j0[1] * qy_[s] + j1[0] * qz_[s] + j1[1] + nq_[s];
#endif
    // Branchless mind update + u64-key argmin (first-min tie-break: equal
    // d2 -> lower node index, matching jnp.argmin).
    unsigned long long bk = ~0ull;
#pragma unroll
    for (int s = 0; s < 4; ++s) {
      float d2 = fmaxf(d2o[s], 0.0f);  // kill cancellation negatives; u64 order valid
      mind_[s] = fminf(mind_[s], d2);
      unsigned long long key =
          ((unsigned long long)__float_as_uint(mind_[s]) << 32) |
          (unsigned int)node_[s];
      key = ((vis >> s) & 1) ? ~0ull : key;
      bk = (key < bk) ? key : bk;
    }
#if HAVE_DPP8
    // Butterfly xor 1,2,4 via DPP8 (pure VALU, no LDS round-trips); after
    // these folds every lane holds its 8-lane group's min.
#define DPP8_FOLD(SEL)                                                        \
    {                                                                         \
      unsigned olo = (unsigned)__builtin_amdgcn_mov_dpp8(                     \
          (int)(unsigned)(bk & 0xFFFFFFFFull), (SEL));                        \
      unsigned ohi = (unsigned)__builtin_amdgcn_mov_dpp8(                     \
          (int)(unsigned)(bk >> 32), (SEL));                                  \
      unsigned long long ok = ((unsigned long long)ohi << 32) | olo;          \
      bk = (ok < bk) ? ok : bk;                                               \
    }
    DPP8_FOLD(DPP8_XOR1)
    DPP8_FOLD(DPP8_XOR2)
    DPP8_FOLD(DPP8_XOR4)
#undef DPP8_FOLD
    const int cur = step & 3;
    if ((lane & 7) == 0) atomicMin(&L.slot[cur], bk);  // 4 group-mins per wave
#else
#pragma unroll
    for (int off = 16; off >= 1; off >>= 1) {
      unsigned long long ok = __shfl_xor(bk, off);
      bk = (ok < bk) ? ok : bk;
    }
    const int cur = step & 3;
    if (lane == 0) atomicMin(&L.slot[cur], bk);
#endif
    if (tid == 0) L.slot[(cur + 2) & 3] = ~0ull;       // prep slot 2 steps ahead
    __syncthreads();                                   // one barrier per step
    unsigned long long wk = L.slot[cur];
    int wj = (int)(wk & 0xFFFFFFFFull);
    if (tid == 0) deaths[step] = __uint_as_float((unsigned int)(wk >> 32));
    Aj = apA[wj];                                      // one ds_load_b64
#if !HAVE_WMMA_F32
    j0 = L.a0[wj]; j1 = L.a1[wj];
#endif
#pragma unroll
    for (int s = 0; s < 4; ++s) vis |= (int)(node_[s] == wj) << s;
  }
  if (tid == 0) deaths[N_PTS - 1] = INF_F;  // pad sorts to the end
  __syncthreads();

  // Bitonic ascending sort of 1024 floats in LDS.
  for (int k = 2; k <= N_PTS; k <<= 1) {
    for (int jj = k >> 1; jj > 0; jj >>= 1) {
      for (int i = tid; i < N_PTS; i += NT) {
        int ix = i ^ jj;
        if (ix > i) {
          float a = deaths[i], b = deaths[ix];
          bool up = (i & k) == 0;
          if ((a > b) == up) { deaths[i] = b; deaths[ix] = a; }
        }
      }
      __syncthreads();
    }
  }
}

__device__ __forceinline__ float block_sum(float v, float* pv8) {
#pragma unroll
  for (int off = 16; off >= 1; off >>= 1) v += __shfl_xor(v, off);
  const int wave = threadIdx.x >> 5;
  const int lane = threadIdx.x & 31;
  __syncthreads();
  if (lane == 0) pv8[wave] = v;
  __syncthreads();
  float s = 0.0f;
#pragma unroll
  for (int w = 0; w < 8; ++w) s += pv8[w];
  return s;  // uniform
}

// One block per cloud (0..15 = gts batches, 16..31 = preds batches):
// Prim + sort, then store sqrt(max(d2, eps)) sorted deaths to workspace.
__global__ void __launch_bounds__(NT) prim32_kernel(const float* __restrict__ gts,
                                                    const float* __restrict__ preds,
                                                    float* __restrict__ wsD) {
  __shared__ PrimLds L;
  __shared__ float deaths[N_PTS];
  const int c = blockIdx.x;
  const float* pts = (c < N_BATCH) ? (gts + (size_t)c * N_PTS * 3)
                                   : (preds + (size_t)(c - N_BATCH) * N_PTS * 3);
  prim_sort_d2(pts, L, deaths);
  for (int i = threadIdx.x; i < N_PTS - 1; i += NT)
    wsD[(size_t)c * N_PTS + i] = sqrtf(fmaxf(deaths[i], EPS_F));
}

// mean_b ( sum_k |dp-dt|^2 )^(1/2)
__global__ void __launch_bounds__(NT) loss_kernel(const float* __restrict__ wsD,
                                                  float* __restrict__ out) {
  __shared__ float pv[8];
  float acc = 0.0f;
  for (int b = 0; b < N_BATCH; ++b) {
    const float* dt = wsD + (size_t)b * N_PTS;
    const float* dp = wsD + (size_t)(N_BATCH + b) * N_PTS;
    float s = 0.0f;
    for (int k = threadIdx.x; k < N_PTS - 1; k += NT) {
      float d = dp[k] - dt[k];
      s += d * d;
    }
    float tot = block_sum(s, pv);
    acc += sqrtf(tot);
  }
  if (threadIdx.x == 0) out[0] = acc / (float)N_BATCH;
}

// Fallback when workspace is too small to hold sorted diagrams: one block
// does everything (slower, still correct and deterministic).
__global__ void __launch_bounds__(NT) mono_kernel(const float* __restrict__ gts,
                                                  const float* __restrict__ preds,
                                                  float* __restrict__ out) {
  __shared__ PrimLds L;
  __shared__ float dA[N_PTS];
  __shared__ float dB[N_PTS];
  float acc = 0.0f;
  for (int b = 0; b < N_BATCH; ++b) {
    prim_sort_d2(gts + (size_t)b * N_PTS * 3, L, dA);
    prim_sort_d2(preds + (size_t)b * N_PTS * 3, L, dB);
    __syncthreads();
    float s = 0.0f;
    for (int k = threadIdx.x; k < N_PTS - 1; k += NT) {
      float d = sqrtf(fmaxf(dB[k], EPS_F)) - sqrtf(fmaxf(dA[k], EPS_F));
      s += d * d;
    }
    float tot = block_sum(s, L.pval);
    acc += sqrtf(tot);
  }
  if (threadIdx.x == 0) out[0] = acc / (float)N_BATCH;
}

extern "C" void kernel_launch(void* const* d_in, const int* in_sizes, int n_in,
                              void* d_out, int out_size, void* d_ws, size_t ws_size,
                              hipStream_t stream) {
  (void)in_sizes; (void)n_in; (void)out_size;
  const float* gts   = (const float*)d_in[0];
  const float* preds = (const float*)d_in[1];
  float* out = (float*)d_out;
  const size_t need = (size_t)(2 * N_BATCH) * N_PTS * sizeof(float);  // 128 KiB
  if (ws_size >= need) {
    float* wsD = (float*)d_ws;
    prim32_kernel<<<32, NT, 0, stream>>>(gts, preds, wsD);
    loss_kernel<<<1, NT, 0, stream>>>(wsD, out);
  } else {
    mono_kernel<<<1, NT, 0, stream>>>(gts, preds, out);
  }
}